// GATClassifier_3255585210603
// MI455X (gfx1250) — compile-verified
//
#include <hip/hip_runtime.h>
#include <hip/hip_bf16.h>
#include <math.h>

// ---- problem constants (match reference) ----
#define N_NODES   100000
#define N_EDGES   1600000
#define ET_EDGES  1700000   // E + N self loops
#define N_GRAPH   512
#define D_INF     128
#define HIDC      64
#define HEADS     4
#define F1        256       // HEADS*HIDC
#define OUTC      2
#define NEG_SLOPE 0.2f
#define BN_EPS    1e-5f

typedef float v2f __attribute__((ext_vector_type(2)));
typedef float v8f __attribute__((ext_vector_type(8)));

#if defined(__has_builtin)
#if __has_builtin(__builtin_amdgcn_wmma_f32_16x16x4_f32)
#define HAVE_WMMA_F32X4 1
#endif
#endif

// ---------------------------------------------------------------------------
// fp32 WMMA GEMM tile: one wave computes a 16x16 tile of OUT = A * B.
// A layout (ISA 16x4 f32): lanes 0-15 hold K=0,1 ; lanes 16-31 hold K=2,3 (M=lane&15)
// B layout assumed mirror of A in K: VGPR0 = (K=half*2, N=lane&15), VGPR1 = K+1
// C/D layout per ISA: VGPR r -> (M = r + 8*(lane>>4), N = lane&15)
// ---------------------------------------------------------------------------

// GEMM1: OUT[n, F1] = emb[x[n], :] @ W1  (fused embedding gather)
__global__ __launch_bounds__(32) void gemm1_wmma(const int* __restrict__ xid,
                                                 const float* __restrict__ emb,
                                                 const float* __restrict__ W1,
                                                 float* __restrict__ out) {
  const int lane = threadIdx.x;
  const int l15  = lane & 15;
  const int half = lane >> 4;
  const int mBase = blockIdx.x * 16;
  const int nBase = blockIdx.y * 16;
#ifdef HAVE_WMMA_F32X4
  const float* arow = emb + (size_t)xid[mBase + l15] * D_INF;
  v8f acc = {0.f, 0.f, 0.f, 0.f, 0.f, 0.f, 0.f, 0.f};
  for (int kk = 0; kk < D_INF; kk += 4) {
    const int ka = kk + half * 2;
    v2f a, b;
    a.x = arow[ka];
    a.y = arow[ka + 1];
    b.x = W1[(size_t)ka * F1 + nBase + l15];
    b.y = W1[(size_t)(ka + 1) * F1 + nBase + l15];
    acc = __builtin_amdgcn_wmma_f32_16x16x4_f32(false, a, false, b, (short)0,
                                                acc, false, false);
  }
#pragma unroll
  for (int r = 0; r < 8; ++r)
    out[(size_t)(mBase + half * 8 + r) * F1 + nBase + l15] = acc[r];
#else
  for (int r = 0; r < 8; ++r) {
    const int m = mBase + half * 8 + r;
    const float* arow = emb + (size_t)xid[m] * D_INF;
    float acc = 0.f;
    for (int k = 0; k < D_INF; ++k)
      acc += arow[k] * W1[(size_t)k * F1 + nBase + l15];
    out[(size_t)m * F1 + nBase + l15] = acc;
  }
#endif
}

// GEMM2: OUT[n, HIDC] = A[n, F1] @ W2
__global__ __launch_bounds__(32) void gemm2_wmma(const float* __restrict__ A,
                                                 const float* __restrict__ W2,
                                                 float* __restrict__ out) {
  const int lane = threadIdx.x;
  const int l15  = lane & 15;
  const int half = lane >> 4;
  const int mBase = blockIdx.x * 16;
  const int nBase = blockIdx.y * 16;
#ifdef HAVE_WMMA_F32X4
  const float* arow = A + (size_t)(mBase + l15) * F1;
  v8f acc = {0.f, 0.f, 0.f, 0.f, 0.f, 0.f, 0.f, 0.f};
  for (int kk = 0; kk < F1; kk += 4) {
    const int ka = kk + half * 2;
    v2f a, b;
    a.x = arow[ka];
    a.y = arow[ka + 1];
    b.x = W2[(size_t)ka * HIDC + nBase + l15];
    b.y = W2[(size_t)(ka + 1) * HIDC + nBase + l15];
    acc = __builtin_amdgcn_wmma_f32_16x16x4_f32(false, a, false, b, (short)0,
                                                acc, false, false);
  }
#pragma unroll
  for (int r = 0; r < 8; ++r)
    out[(size_t)(mBase + half * 8 + r) * HIDC + nBase + l15] = acc[r];
#else
  for (int r = 0; r < 8; ++r) {
    const int m = mBase + half * 8 + r;
    const float* arow = A + (size_t)m * F1;
    float acc = 0.f;
    for (int k = 0; k < F1; ++k)
      acc += arow[k] * W2[(size_t)k * HIDC + nBase + l15];
    out[(size_t)m * HIDC + nBase + l15] = acc;
  }
#endif
}

// ---------------------------------------------------------------------------
__global__ void fill_f32(float* __restrict__ p, int n, float v) {
  int i = blockIdx.x * blockDim.x + threadIdx.x;
  if (i < n) p[i] = v;
}

// attention coefficients: a[n,h] = sum_c H[n, h*C + c] * att[h, c]
template <int H, int C>
__global__ void attncoef_kernel(const float* __restrict__ Hbuf,
                                const float* __restrict__ att_s,
                                const float* __restrict__ att_d,
                                float* __restrict__ as_, float* __restrict__ ad_) {
  int i = blockIdx.x * blockDim.x + threadIdx.x;
  if (i >= N_NODES * H) return;
  int n = i / H, hh = i % H;
  const float* hp = Hbuf + (size_t)n * (H * C) + hh * C;
  float ss = 0.f, dd = 0.f;
  for (int c = 0; c < C; ++c) {
    float hv = hp[c];
    ss += hv * att_s[hh * C + c];
    dd += hv * att_d[hh * C + c];
  }
  as_[i] = ss;
  ad_[i] = dd;
}

__device__ inline void atomicMaxFloat(float* addr, float val) {
  if (val >= 0.f)
    atomicMax((int*)addr, __float_as_int(val));
  else
    atomicMin((unsigned int*)addr, (unsigned int)__float_as_int(val));
}

__device__ inline float leaky(float v) { return v > 0.f ? v : NEG_SLOPE * v; }

// segment max of leaky_relu(a_src[src]+a_dst[dst]) over dst
template <int H>
__global__ void edge_max_kernel(const int* __restrict__ srcp, const int* __restrict__ dstp,
                                const float* __restrict__ as_, const float* __restrict__ ad_,
                                float* __restrict__ m) {
  long long idx = (long long)blockIdx.x * blockDim.x + threadIdx.x;
  if (idx >= (long long)ET_EDGES * H) return;
  int e = (int)(idx / H), hh = (int)(idx % H);
  int s = e < N_EDGES ? srcp[e] : e - N_EDGES;
  int d = e < N_EDGES ? dstp[e] : e - N_EDGES;
  atomicMaxFloat(&m[d * H + hh], leaky(as_[s * H + hh] + ad_[d * H + hh]));
}

// segment sum of exp(e - m[dst])
template <int H>
__global__ void edge_sum_kernel(const int* __restrict__ srcp, const int* __restrict__ dstp,
                                const float* __restrict__ as_, const float* __restrict__ ad_,
                                const float* __restrict__ m, float* __restrict__ sbuf) {
  long long idx = (long long)blockIdx.x * blockDim.x + threadIdx.x;
  if (idx >= (long long)ET_EDGES * H) return;
  int e = (int)(idx / H), hh = (int)(idx % H);
  int s = e < N_EDGES ? srcp[e] : e - N_EDGES;
  int d = e < N_EDGES ? dstp[e] : e - N_EDGES;
  float v = leaky(as_[s * H + hh] + ad_[d * H + hh]);
  atomicAdd(&sbuf[d * H + hh], __expf(v - m[d * H + hh]));
}

// layer1 message scatter: one wave per edge, 256 channels (4 heads x 64)
__global__ __launch_bounds__(256) void edge_scatter1_kernel(
    const int* __restrict__ srcp, const int* __restrict__ dstp,
    const float* __restrict__ as_, const float* __restrict__ ad_,
    const float* __restrict__ m, const float* __restrict__ sbuf,
    const float* __restrict__ H1, float* __restrict__ agg) {
  int lane = threadIdx.x & 31;
  int e = blockIdx.x * (blockDim.x >> 5) + (threadIdx.x >> 5);
  if (e >= ET_EDGES) return;
  int s = e < N_EDGES ? srcp[e] : e - N_EDGES;
  int d = e < N_EDGES ? dstp[e] : e - N_EDGES;
  const float* hs = H1 + (size_t)s * F1;
  float* ag = agg + (size_t)d * F1;
#pragma unroll
  for (int j = 0; j < 8; ++j) {
    int cg = j * 32 + lane;
    int hh = cg >> 6;
    float v = leaky(as_[s * HEADS + hh] + ad_[d * HEADS + hh]);
    float alpha = __expf(v - m[d * HEADS + hh]) / sbuf[d * HEADS + hh];
    atomicAdd(&ag[cg], hs[cg] * alpha);
  }
}

// layer2 message scatter: one wave per edge, 64 channels, 1 head
__global__ __launch_bounds__(256) void edge_scatter2_kernel(
    const int* __restrict__ srcp, const int* __restrict__ dstp,
    const float* __restrict__ as_, const float* __restrict__ ad_,
    const float* __restrict__ m, const float* __restrict__ sbuf,
    const float* __restrict__ H2, float* __restrict__ agg) {
  int lane = threadIdx.x & 31;
  int e = blockIdx.x * (blockDim.x >> 5) + (threadIdx.x >> 5);
  if (e >= ET_EDGES) return;
  int s = e < N_EDGES ? srcp[e] : e - N_EDGES;
  int d = e < N_EDGES ? dstp[e] : e - N_EDGES;
  float v = leaky(as_[s] + ad_[d]);
  float alpha = __expf(v - m[d]) / sbuf[d];
  const float* hs = H2 + (size_t)s * HIDC;
  float* ag = agg + (size_t)d * HIDC;
#pragma unroll
  for (int j = 0; j < 2; ++j) {
    int cg = j * 32 + lane;
    atomicAdd(&ag[cg], hs[cg] * alpha);
  }
}

// bias + BN(eval) + ELU, in place (layer1: C = 256)
__global__ void post1_kernel(float* __restrict__ buf, const float* __restrict__ bias,
                             const float* __restrict__ gamma, const float* __restrict__ beta,
                             const float* __restrict__ mean, const float* __restrict__ var) {
  int i = blockIdx.x * blockDim.x + threadIdx.x;
  if (i >= N_NODES * F1) return;
  int c = i & (F1 - 1);
  float v = buf[i] + bias[c];
  v = (v - mean[c]) * rsqrtf(var[c] + BN_EPS) * gamma[c] + beta[c];
  buf[i] = v > 0.f ? v : expm1f(v);
}

// bias + BN + ELU + mean-pool scatter (layer2: C = 64)
__global__ void post2_pool_kernel(const float* __restrict__ agg, const int* __restrict__ batch,
                                  const float* __restrict__ bias, const float* __restrict__ gamma,
                                  const float* __restrict__ beta, const float* __restrict__ mean,
                                  const float* __restrict__ var, float* __restrict__ pooled) {
  int i = blockIdx.x * blockDim.x + threadIdx.x;
  if (i >= N_NODES * HIDC) return;
  int n = i >> 6, c = i & 63;
  float v = agg[i] + bias[c];
  v = (v - mean[c]) * rsqrtf(var[c] + BN_EPS) * gamma[c] + beta[c];
  v = v > 0.f ? v : expm1f(v);
  atomicAdd(&pooled[(size_t)batch[n] * HIDC + c], v);
}

__global__ void count_kernel(const int* __restrict__ batch, float* __restrict__ cnt) {
  int i = blockIdx.x * blockDim.x + threadIdx.x;
  if (i < N_NODES) atomicAdd(&cnt[batch[i]], 1.f);
}

// final per-graph MLP: relu(pooled/cnt @ lin1 + b1) @ lin2 + b2
__global__ __launch_bounds__(64) void mlp_kernel(const float* __restrict__ pooled,
                                                 const float* __restrict__ cnt,
                                                 const float* __restrict__ lin1_w,
                                                 const float* __restrict__ lin1_b,
                                                 const float* __restrict__ lin2_w,
                                                 const float* __restrict__ lin2_b,
                                                 float* __restrict__ out) {
  __shared__ float p[HIDC];
  __shared__ float z[HIDC];
  int g = blockIdx.x, t = threadIdx.x;
  float c = cnt[g];
  c = c > 1.f ? c : 1.f;
  p[t] = pooled[(size_t)g * HIDC + t] / c;
  __syncthreads();
  float acc = lin1_b[t];
  for (int i = 0; i < HIDC; ++i) acc += p[i] * lin1_w[i * HIDC + t];
  z[t] = acc > 0.f ? acc : 0.f;
  __syncthreads();
  if (t < OUTC) {
    float o = lin2_b[t];
    for (int j = 0; j < HIDC; ++j) o += z[j] * lin2_w[j * OUTC + t];
    out[g * OUTC + t] = o;
  }
}

// ---------------------------------------------------------------------------
extern "C" void kernel_launch(void* const* d_in, const int* in_sizes, int n_in,
                              void* d_out, int out_size, void* d_ws, size_t ws_size,
                              hipStream_t stream) {
  (void)in_sizes; (void)n_in; (void)out_size; (void)ws_size;
  const int*   x       = (const int*)d_in[0];
  const int*   ei      = (const int*)d_in[1];   // [2, E] row-major
  const int*   batch   = (const int*)d_in[2];
  const float* emb     = (const float*)d_in[3];
  const float* W1      = (const float*)d_in[4];
  const float* attS1   = (const float*)d_in[5];
  const float* attD1   = (const float*)d_in[6];
  const float* bias1   = (const float*)d_in[7];
  const float* gamma1  = (const float*)d_in[8];
  const float* beta1   = (const float*)d_in[9];
  const float* mean1   = (const float*)d_in[10];
  const float* var1    = (const float*)d_in[11];
  const float* W2      = (const float*)d_in[12];
  const float* attS2   = (const float*)d_in[13];
  const float* attD2   = (const float*)d_in[14];
  const float* bias2   = (const float*)d_in[15];
  const float* gamma2  = (const float*)d_in[16];
  const float* beta2   = (const float*)d_in[17];
  const float* mean2   = (const float*)d_in[18];
  const float* var2    = (const float*)d_in[19];
  const float* lin1_w  = (const float*)d_in[20];
  const float* lin1_b  = (const float*)d_in[21];
  const float* lin2_w  = (const float*)d_in[22];
  const float* lin2_b  = (const float*)d_in[23];
  float* out = (float*)d_out;

  const int* srcp = ei;
  const int* dstp = ei + N_EDGES;

  // ---- workspace layout (floats) ----
  float* ws    = (float*)d_ws;
  float* bufA  = ws;                               // N*256 : H1raw, later H2raw/agg2
  float* bufB  = bufA + (size_t)N_NODES * F1;      // N*256 : agg1 -> h1post (in place)
  float* asrc1 = bufB + (size_t)N_NODES * F1;      // N*4
  float* adst1 = asrc1 + N_NODES * HEADS;          // N*4
  float* m1    = adst1 + N_NODES * HEADS;          // N*4
  float* s1    = m1 + N_NODES * HEADS;             // N*4
  float* pooled = s1 + N_NODES * HEADS;            // G*64
  float* cnt    = pooled + N_GRAPH * HIDC;         // G
  // layer-2 aliases (safe by stream ordering)
  float* H2   = bufA;                              // N*64
  float* agg2 = bufA + (size_t)N_NODES * HIDC;     // N*64
  float* asrc2 = asrc1; float* adst2 = adst1; float* m2 = m1; float* s2 = s1;

  const int BT = 256;
  const int scatterBlocks = (ET_EDGES + 7) / 8;  // 8 waves (edges) per 256-thread block

  // ===== layer 1 =====
  gemm1_wmma<<<dim3(N_NODES / 16, F1 / 16), 32, 0, stream>>>(x, emb, W1, bufA);
  attncoef_kernel<HEADS, HIDC><<<(N_NODES * HEADS + BT - 1) / BT, BT, 0, stream>>>(
      bufA, attS1, attD1, asrc1, adst1);
  fill_f32<<<(N_NODES * HEADS + BT - 1) / BT, BT, 0, stream>>>(m1, N_NODES * HEADS, -INFINITY);
  fill_f32<<<(N_NODES * HEADS + BT - 1) / BT, BT, 0, stream>>>(s1, N_NODES * HEADS, 0.f);
  fill_f32<<<(N_NODES * F1 + BT - 1) / BT, BT, 0, stream>>>(bufB, N_NODES * F1, 0.f);
  edge_max_kernel<HEADS><<<(ET_EDGES * HEADS + BT - 1) / BT, BT, 0, stream>>>(
      srcp, dstp, asrc1, adst1, m1);
  edge_sum_kernel<HEADS><<<(ET_EDGES * HEADS + BT - 1) / BT, BT, 0, stream>>>(
      srcp, dstp, asrc1, adst1, m1, s1);
  edge_scatter1_kernel<<<scatterBlocks, BT, 0, stream>>>(
      srcp, dstp, asrc1, adst1, m1, s1, bufA, bufB);
  post1_kernel<<<(N_NODES * F1 + BT - 1) / BT, BT, 0, stream>>>(
      bufB, bias1, gamma1, beta1, mean1, var1);

  // ===== layer 2 =====
  gemm2_wmma<<<dim3(N_NODES / 16, HIDC / 16), 32, 0, stream>>>(bufB, W2, H2);
  attncoef_kernel<1, HIDC><<<(N_NODES + BT - 1) / BT, BT, 0, stream>>>(
      H2, attS2, attD2, asrc2, adst2);
  fill_f32<<<(N_NODES + BT - 1) / BT, BT, 0, stream>>>(m2, N_NODES, -INFINITY);
  fill_f32<<<(N_NODES + BT - 1) / BT, BT, 0, stream>>>(s2, N_NODES, 0.f);
  fill_f32<<<(N_NODES * HIDC + BT - 1) / BT, BT, 0, stream>>>(agg2, N_NODES * HIDC, 0.f);
  edge_max_kernel<1><<<(ET_EDGES + BT - 1) / BT, BT, 0, stream>>>(
      srcp, dstp, asrc2, adst2, m2);
  edge_sum_kernel<1><<<(ET_EDGES + BT - 1) / BT, BT, 0, stream>>>(
      srcp, dstp, asrc2, adst2, m2, s2);
  edge_scatter2_kernel<<<scatterBlocks, BT, 0, stream>>>(
      srcp, dstp, asrc2, adst2, m2, s2, H2, agg2);

  // ===== pool + MLP =====
  fill_f32<<<(N_GRAPH * HIDC + BT - 1) / BT, BT, 0, stream>>>(pooled, N_GRAPH * HIDC, 0.f);
  fill_f32<<<(N_GRAPH + BT - 1) / BT, BT, 0, stream>>>(cnt, N_GRAPH, 0.f);
  count_kernel<<<(N_NODES + BT - 1) / BT, BT, 0, stream>>>(batch, cnt);
  post2_pool_kernel<<<(N_NODES * HIDC + BT - 1) / BT, BT, 0, stream>>>(
      agg2, batch, bias2, gamma2, beta2, mean2, var2, pooled);
  mlp_kernel<<<N_GRAPH, HIDC, 0, stream>>>(pooled, cnt, lin1_w, lin1_b, lin2_w, lin2_b, out);
}